// TopicGraphModel_9560597201474
// MI455X (gfx1250) — compile-verified
//
#include <hip/hip_runtime.h>

typedef float v2f __attribute__((ext_vector_type(2)));
typedef float v8f __attribute__((ext_vector_type(8)));

#define IN_F  128
#define HID_F 64
#define CLS_F 32

// ---------------- degree / norm ----------------
__global__ __launch_bounds__(256) void init_deg_k(float* __restrict__ deg, int n) {
    int i = blockIdx.x * 256 + threadIdx.x;
    if (i < n) deg[i] = 1.0f;                       // self-loop contributes 1
}

__global__ __launch_bounds__(256) void edge_deg_k(const int* __restrict__ dst,
                                                  float* __restrict__ deg, int E) {
    int e = blockIdx.x * 256 + threadIdx.x;
    if (e < E)
        __hip_atomic_fetch_add(&deg[dst[e]], 1.0f, __ATOMIC_RELAXED,
                               __HIP_MEMORY_SCOPE_AGENT);
}

__global__ __launch_bounds__(256) void dinv_k(float* __restrict__ deg, int n) {
    int i = blockIdx.x * 256 + threadIdx.x;
    if (i < n) {
        float d = deg[i];
        deg[i] = (d > 0.0f) ? rsqrtf(d) : 0.0f;    // in-place: buffer becomes dinv
    }
}

// ---------------- layer 1 GEMM: h = x @ W1, a1 = h * dinv^2 (self loop) ----------------
__global__ __launch_bounds__(256) void gemm1_wmma(const float* __restrict__ x,
                                                  const float* __restrict__ W1,
                                                  const float* __restrict__ dinv,
                                                  float* __restrict__ h,
                                                  float* __restrict__ a1,
                                                  int nRowTiles) {
    __shared__ float ldsW[IN_F * HID_F];            // 32 KB
    for (int i = threadIdx.x; i < IN_F * HID_F; i += 256) ldsW[i] = W1[i];
    __syncthreads();

    int wave = (blockIdx.x * 256 + threadIdx.x) >> 5;
    int lane = threadIdx.x & 31;
    int nTiles = nRowTiles * (HID_F / 16);
    if (wave >= nTiles) return;                     // wave-uniform: EXEC stays all-ones

    int rt   = wave / (HID_F / 16);
    int ct   = wave % (HID_F / 16);
    int half = lane >> 4;                           // A: K pair select; D: M-half select
    int l    = lane & 15;
    int row  = rt * 16 + l;
    int col  = ct * 16 + l;

    const float* xrow = x + (size_t)row * IN_F;
    v8f acc = {};
#pragma unroll
    for (int kb = 0; kb < IN_F; kb += 4) {
        int ka = kb + half * 2;
        v2f a = *(const v2f*)(xrow + ka);           // A[M=row][K=ka..ka+1]
        v2f b;
        b.x = ldsW[ka * HID_F + col];               // B[K=ka][N=col]
        b.y = ldsW[(ka + 1) * HID_F + col];
        acc = __builtin_amdgcn_wmma_f32_16x16x4_f32(false, a, false, b,
                                                    (short)0, acc, false, false);
    }

    int mbase = rt * 16 + half * 8;                 // D: lanes16-31 hold M=8..15
#pragma unroll
    for (int r = 0; r < 8; ++r) {
        int m    = mbase + r;
        float v  = acc[r];
        float di = dinv[m];
        h[(size_t)m * HID_F + col]  = v;
        a1[(size_t)m * HID_F + col] = v * di * di;  // self-loop term of segment_sum
    }
}

// ---------------- layer 1 edge scatter: one wave per edge (64 feats, float2/lane) ----------------
__global__ __launch_bounds__(256) void scatter1_k(const int* __restrict__ src,
                                                  const int* __restrict__ dst,
                                                  const float* __restrict__ h,
                                                  const float* __restrict__ dinv,
                                                  float* __restrict__ a1, int E) {
    long long t = (long long)blockIdx.x * 256 + threadIdx.x;
    int e = (int)(t >> 5);
    if (e >= E) return;
    int fp = ((int)t & 31) * 2;
    int s = src[e], d = dst[e];
    float norm = dinv[s] * dinv[d];
    v2f hv = *(const v2f*)(h + (size_t)s * HID_F + fp);
    float* ap = a1 + (size_t)d * HID_F + fp;
    __hip_atomic_fetch_add(ap,     hv.x * norm, __ATOMIC_RELAXED, __HIP_MEMORY_SCOPE_AGENT);
    __hip_atomic_fetch_add(ap + 1, hv.y * norm, __ATOMIC_RELAXED, __HIP_MEMORY_SCOPE_AGENT);
}

// ---------------- layer 2 GEMM: g2 = relu(a1+b1) @ W2, out = g2*dinv^2 + b2 ----------------
__global__ __launch_bounds__(256) void gemm2_wmma(const float* __restrict__ a1,
                                                  const float* __restrict__ b1,
                                                  const float* __restrict__ W2,
                                                  const float* __restrict__ b2,
                                                  const float* __restrict__ dinv,
                                                  float* __restrict__ g2,
                                                  float* __restrict__ out,
                                                  int nRowTiles) {
    __shared__ float ldsW[HID_F * CLS_F];           // 8 KB
    for (int i = threadIdx.x; i < HID_F * CLS_F; i += 256) ldsW[i] = W2[i];
    __syncthreads();

    int wave = (blockIdx.x * 256 + threadIdx.x) >> 5;
    int lane = threadIdx.x & 31;
    int nTiles = nRowTiles * (CLS_F / 16);
    if (wave >= nTiles) return;                     // wave-uniform

    int rt   = wave / (CLS_F / 16);
    int ct   = wave % (CLS_F / 16);
    int half = lane >> 4;
    int l    = lane & 15;
    int row  = rt * 16 + l;
    int col  = ct * 16 + l;

    const float* arow = a1 + (size_t)row * HID_F;
    v8f acc = {};
#pragma unroll
    for (int kb = 0; kb < HID_F; kb += 4) {
        int ka = kb + half * 2;
        v2f a;                                       // fused bias + relu on A load
        a.x = fmaxf(arow[ka]     + b1[ka],     0.0f);
        a.y = fmaxf(arow[ka + 1] + b1[ka + 1], 0.0f);
        v2f b;
        b.x = ldsW[ka * CLS_F + col];
        b.y = ldsW[(ka + 1) * CLS_F + col];
        acc = __builtin_amdgcn_wmma_f32_16x16x4_f32(false, a, false, b,
                                                    (short)0, acc, false, false);
    }

    int mbase = rt * 16 + half * 8;
#pragma unroll
    for (int r = 0; r < 8; ++r) {
        int m    = mbase + r;
        float v  = acc[r];
        float di = dinv[m];
        g2[(size_t)m * CLS_F + col]  = v;
        out[(size_t)m * CLS_F + col] = v * di * di + b2[col]; // self loop + bias
    }
}

// ---------------- layer 2 edge scatter: 16 lanes per edge (32 feats, float2/lane) ----------------
__global__ __launch_bounds__(256) void scatter2_k(const int* __restrict__ src,
                                                  const int* __restrict__ dst,
                                                  const float* __restrict__ g2,
                                                  const float* __restrict__ dinv,
                                                  float* __restrict__ out, int E) {
    long long t = (long long)blockIdx.x * 256 + threadIdx.x;
    int e = (int)(t >> 4);
    if (e >= E) return;
    int fp = ((int)t & 15) * 2;
    int s = src[e], d = dst[e];
    float norm = dinv[s] * dinv[d];
    v2f gv = *(const v2f*)(g2 + (size_t)s * CLS_F + fp);
    float* op = out + (size_t)d * CLS_F + fp;
    __hip_atomic_fetch_add(op,     gv.x * norm, __ATOMIC_RELAXED, __HIP_MEMORY_SCOPE_AGENT);
    __hip_atomic_fetch_add(op + 1, gv.y * norm, __ATOMIC_RELAXED, __HIP_MEMORY_SCOPE_AGENT);
}

extern "C" void kernel_launch(void* const* d_in, const int* in_sizes, int n_in,
                              void* d_out, int out_size, void* d_ws, size_t ws_size,
                              hipStream_t stream) {
    const float* x  = (const float*)d_in[0];
    const int*   ei = (const int*)  d_in[1];
    const float* W1 = (const float*)d_in[2];
    const float* b1 = (const float*)d_in[3];
    const float* W2 = (const float*)d_in[4];
    const float* b2 = (const float*)d_in[5];
    float* out = (float*)d_out;

    int n = in_sizes[0] / IN_F;       // 100000
    int E = in_sizes[1] / 2;          // 1600000
    const int* src = ei;              // edge_index[0, :]
    const int* dst = ei + E;          // edge_index[1, :]

    // workspace layout (256B aligned): dinv | h | a1 | g2  (~64.5 MB)
    char*  ws       = (char*)d_ws;
    size_t off_dinv = 0;
    size_t off_h    = (off_dinv + (size_t)n * 4            + 255) & ~(size_t)255;
    size_t off_a1   = (off_h    + (size_t)n * HID_F * 4    + 255) & ~(size_t)255;
    size_t off_g2   = (off_a1   + (size_t)n * HID_F * 4    + 255) & ~(size_t)255;
    float* dinv = (float*)(ws + off_dinv);
    float* h    = (float*)(ws + off_h);
    float* a1   = (float*)(ws + off_a1);
    float* g2   = (float*)(ws + off_g2);

    int nb = (n + 255) / 256;
    init_deg_k<<<nb, 256, 0, stream>>>(dinv, n);
    edge_deg_k<<<(E + 255) / 256, 256, 0, stream>>>(dst, dinv, E);
    dinv_k<<<nb, 256, 0, stream>>>(dinv, n);

    int nRowTiles = (n + 15) / 16;                       // 6250 (n is a multiple of 16)
    int waves1 = nRowTiles * (HID_F / 16);               // 25000 waves
    gemm1_wmma<<<(waves1 + 7) / 8, 256, 0, stream>>>(x, W1, dinv, h, a1, nRowTiles);

    long long t1 = (long long)E * 32;                    // one wave per edge
    scatter1_k<<<(int)((t1 + 255) / 256), 256, 0, stream>>>(src, dst, h, dinv, a1, E);

    int waves2 = nRowTiles * (CLS_F / 16);               // 12500 waves
    gemm2_wmma<<<(waves2 + 7) / 8, 256, 0, stream>>>(a1, b1, W2, b2, dinv, g2, out, nRowTiles);

    long long t2 = (long long)E * 16;                    // half-wave per edge
    scatter2_k<<<(int)((t2 + 255) / 256), 256, 0, stream>>>(src, dst, g2, dinv, out, E);
}